// FlatHashConvNNUE_47519518163395
// MI455X (gfx1250) — compile-verified
//
#include <hip/hip_runtime.h>
#include <hip/hip_bf16.h>

typedef __attribute__((ext_vector_type(2))) float v2f;
typedef __attribute__((ext_vector_type(8))) float v8f;
typedef __attribute__((ext_vector_type(2))) short v2s;

#define QMAXF 0.9921875f   /* 127/128 */

__device__ __forceinline__ float quant_w8(float w) {
  // fake_quant(w, 128, bits=8): clip(round(w*128), -128, 127)/128
  float y = rintf(w * 128.f);
  y = fminf(fmaxf(y, -128.f), 127.f);
  return y * (1.f / 128.f);
}
__device__ __forceinline__ float quant_b32(float b) {
  // fake_quant(b, 128*128, bits=32): round on 1/16384 grid (clamp never hit)
  return rintf(b * 16384.f) * (1.f / 16384.f);
}
__device__ __forceinline__ int quant_feat(float x) {
  // clip(x, -1, 127/128) then round-half-even to int8 grid (exact integer)
  float xc = fminf(fmaxf(x, -1.f), QMAXF);
  return __float2int_rn(xc * 128.f);
}
__device__ __forceinline__ v2s shflxor_pk(v2s v, int m) {
  int t = __shfl_xor(__builtin_bit_cast(int, v), m);
  return __builtin_bit_cast(v2s, t);
}

// ---- Phase 1 (shared): 15-bit row masks + 169 perfect-hash window indices ----
__device__ __forceinline__ void build_indices(const int* __restrict__ board,
                                              int* __restrict__ sIdx,
                                              int lane, int wave, int bBase, int Btot)
{
  for (int q = 0; q < 4; ++q) {
    const int slot = wave * 4 + q;
    const int b    = bBase + slot;
    int mask = 0;
    if (lane < 30 && b < Btot) {
      const int c = lane / 15, y = lane % 15;
      const int* row = board + (size_t)b * 450 + c * 225 + y * 15;
      #pragma unroll
      for (int x = 0; x < 15; ++x) mask |= (row[x] & 1) << x;
    }
    #pragma unroll
    for (int chunk = 0; chunk < 6; ++chunk) {
      const int w = chunk * 32 + lane;
      const int i = w / 13, j = w - i * 13;
      int idx = 0;
      #pragma unroll
      for (int r = 0; r < 3; ++r) {
        const int m0 = __shfl(mask, i + r);        // channel 0, row i+r
        const int m1 = __shfl(mask, 15 + i + r);   // channel 1, row i+r
        idx |= ((m0 >> j) & 7) << (r * 3);
        idx |= ((m1 >> j) & 7) << (9 + r * 3);
      }
      if (w < 169) sIdx[slot * 176 + w] = idx;
    }
  }
}

// ---- Phase 3 (shared): quantized 3-layer MLP via V_WMMA_F32_16X16X4_F32 ----
__device__ __forceinline__ void mlp_wmma(float* __restrict__ sV0, float* __restrict__ sV1,
                                         const float* __restrict__ w1, const float* __restrict__ b1,
                                         const float* __restrict__ w2, const float* __restrict__ b2,
                                         const float* __restrict__ w3, const float* __restrict__ b3,
                                         float* __restrict__ vout,
                                         int lane, int bBase, int Btot)
{
  const int hi = lane >> 4;     // A: K pair select; C: row +8
  const int ln = lane & 15;     // A: row M; B/C: col N

  auto run_layer = [&](const float* __restrict__ Vin, float* __restrict__ Vout,
                       const float* __restrict__ W, const float* __restrict__ Bv,
                       int nValid, float clipLo, int nTiles, bool finalStore) {
    for (int t = 0; t < nTiles; ++t) {
      const int n0 = t * 16;
      const int n  = n0 + ln;
      const float cb = (n < nValid) ? quant_b32(Bv[n]) : 0.f;
      v8f c;
      #pragma unroll
      for (int r = 0; r < 8; ++r) c[r] = cb;
      #pragma unroll
      for (int k = 0; k < 32; k += 4) {
        const int kk = k + hi * 2;
        v2f a, bf;
        float a0 = Vin[ln * 33 + kk];
        float a1 = Vin[ln * 33 + kk + 1];
        a[0] = fminf(fmaxf(a0, clipLo), QMAXF);
        a[1] = fminf(fmaxf(a1, clipLo), QMAXF);
        bf[0] = (n < nValid) ? quant_w8(W[n * 32 + kk])     : 0.f;   // B[k][n]=Wq[n][k]
        bf[1] = (n < nValid) ? quant_w8(W[n * 32 + kk + 1]) : 0.f;
        c = __builtin_amdgcn_wmma_f32_16x16x4_f32(false, a, false, bf,
                                                  (short)0, c, false, false);
      }
      if (!finalStore) {
        #pragma unroll
        for (int r = 0; r < 8; ++r) Vout[(r + hi * 8) * 33 + n] = c[r];
      } else if (ln < 3) {
        #pragma unroll
        for (int r = 0; r < 8; ++r) {
          const int bb = bBase + r + hi * 8;
          if (bb < Btot) vout[(size_t)bb * 3 + ln] = c[r];
        }
      }
    }
  };

  run_layer(sV0, sV1, w1, b1, 32, -1.f, 2, false);   // clip [-1, 127/128]
  run_layer(sV1, sV0, w2, b2, 32,  0.f, 2, false);   // clip [ 0, 127/128]
  run_layer(sV0, nullptr, w3, b3, 3, 0.f, 1, true);  // 3 outputs, N padded
}

// ---- Prep: quantize embedding table to packed int8 (4x less gather traffic) ----
__global__ void quant_emb_kernel(const float* __restrict__ emb,
                                 unsigned* __restrict__ q, int n4)
{
  for (int i = blockIdx.x * blockDim.x + threadIdx.x; i < n4;
       i += gridDim.x * blockDim.x) {
    const float4 x = ((const float4*)emb)[i];
    const unsigned b0 = (unsigned)quant_feat(x.x) & 255u;
    const unsigned b1 = (unsigned)quant_feat(x.y) & 255u;
    const unsigned b2 = (unsigned)quant_feat(x.z) & 255u;
    const unsigned b3 = (unsigned)quant_feat(x.w) & 255u;
    q[i] = b0 | (b1 << 8) | (b2 << 16) | (b3 << 24);
  }
}

// ---- Main kernel, int8-table path: 32-B rows, packed i16 accumulation ----
__global__ __launch_bounds__(128) void
nnue_fwd_q8(const int* __restrict__ board, const unsigned* __restrict__ qemb,
            const float* __restrict__ w1, const float* __restrict__ b1,
            const float* __restrict__ w2, const float* __restrict__ b2,
            const float* __restrict__ w3, const float* __restrict__ b3,
            float* __restrict__ vout, int Btot)
{
  __shared__ int   sIdx[16 * 176];
  __shared__ float sV0[16 * 33];
  __shared__ float sV1[16 * 33];

  const int tid   = threadIdx.x;
  const int lane  = tid & 31;
  const int wave  = tid >> 5;
  const int bBase = blockIdx.x * 16;

  build_indices(board, sIdx, lane, wave, bBase, Btot);

  // wave gathers 8 rows/iter; 4 lanes per 32-B row, uint2 (8 dims) per lane
  const int g  = lane >> 2;        // row slot 0..7
  const int dg = (lane & 3) * 8;   // dim base 0,8,16,24
  for (int q = 0; q < 4; ++q) {
    const int slot = wave * 4 + q;
    const int b    = bBase + slot;
    v2s a0 = {0, 0}, a1 = {0, 0}, a2 = {0, 0}, a3 = {0, 0};
    if (b < Btot) {
      const int* idxRow = &sIdx[slot * 176];
      for (int it = 0; it < 22; ++it) {        // ceil(169/8)
        const int w = it * 8 + g;
        if (w < 169) {
          const int idx = idxRow[w];
          if (w + 16 < 169) {                  // prefetch 2 iterations ahead
            const int idx2 = idxRow[w + 16];
            __builtin_prefetch(qemb + (size_t)idx2 * 8 + (lane & 3) * 2, 0, 0);
          }
          const uint2 d = *(const uint2*)(qemb + (size_t)idx * 8 + (lane & 3) * 2);
          // packed sign-extend: v_pk_ashrrev_i16 + v_pk_add_i16
          a0 += (__builtin_bit_cast(v2s, d.x << 8) >> 8);  // dims +0,+2
          a1 += (__builtin_bit_cast(v2s, d.x) >> 8);       // dims +1,+3
          a2 += (__builtin_bit_cast(v2s, d.y << 8) >> 8);  // dims +4,+6
          a3 += (__builtin_bit_cast(v2s, d.y) >> 8);       // dims +5,+7
        }
      }
    }
    // fold the 8 row-slots: lanes sharing (lane&3); i16 sums stay in range
    #pragma unroll
    for (int m = 4; m <= 16; m <<= 1) {
      a0 += shflxor_pk(a0, m); a1 += shflxor_pk(a1, m);
      a2 += shflxor_pk(a2, m); a3 += shflxor_pk(a3, m);
    }
    if (lane < 4) {
      const float s = 1.f / 128.f;
      float* vr = &sV0[slot * 33 + dg];
      vr[0] = a0[0] * s; vr[1] = a1[0] * s; vr[2] = a0[1] * s; vr[3] = a1[1] * s;
      vr[4] = a2[0] * s; vr[5] = a3[0] * s; vr[6] = a2[1] * s; vr[7] = a3[1] * s;
    }
  }

  __syncthreads();
  if (tid < 32)
    mlp_wmma(sV0, sV1, w1, b1, w2, b2, w3, b3, vout, lane, bBase, Btot);
}

// ---- Fallback path (ws too small): gather f32 rows, quantize on the fly ----
__global__ __launch_bounds__(128) void
nnue_fwd_f32(const int* __restrict__ board, const float* __restrict__ emb,
             const float* __restrict__ w1, const float* __restrict__ b1,
             const float* __restrict__ w2, const float* __restrict__ b2,
             const float* __restrict__ w3, const float* __restrict__ b3,
             float* __restrict__ vout, int Btot)
{
  __shared__ int   sIdx[16 * 176];
  __shared__ float sV0[16 * 33];
  __shared__ float sV1[16 * 33];

  const int tid   = threadIdx.x;
  const int lane  = tid & 31;
  const int wave  = tid >> 5;
  const int bBase = blockIdx.x * 16;

  build_indices(board, sIdx, lane, wave, bBase, Btot);

  const int g  = lane >> 3;        // row slot 0..3
  const int dg = (lane & 7) * 4;   // dim base
  for (int q = 0; q < 4; ++q) {
    const int slot = wave * 4 + q;
    const int b    = bBase + slot;
    int acc0 = 0, acc1 = 0, acc2 = 0, acc3 = 0;
    if (b < Btot) {
      const int* idxRow = &sIdx[slot * 176];
      for (int it = 0; it < 43; ++it) {
        const int w = it * 4 + g;
        if (w < 169) {
          const int idx = idxRow[w];
          if (w + 8 < 169) {
            const int idx2 = idxRow[w + 8];
            __builtin_prefetch(emb + (size_t)idx2 * 32 + dg, 0, 0);
          }
          const float4 x = *(const float4*)(emb + (size_t)idx * 32 + dg);
          acc0 += quant_feat(x.x); acc1 += quant_feat(x.y);
          acc2 += quant_feat(x.z); acc3 += quant_feat(x.w);
        }
      }
    }
    acc0 += __shfl_xor(acc0, 8); acc0 += __shfl_xor(acc0, 16);
    acc1 += __shfl_xor(acc1, 8); acc1 += __shfl_xor(acc1, 16);
    acc2 += __shfl_xor(acc2, 8); acc2 += __shfl_xor(acc2, 16);
    acc3 += __shfl_xor(acc3, 8); acc3 += __shfl_xor(acc3, 16);
    if (lane < 8) {
      const float s = 1.f / 128.f;
      float* vr = &sV0[slot * 33 + lane * 4];
      vr[0] = acc0 * s; vr[1] = acc1 * s; vr[2] = acc2 * s; vr[3] = acc3 * s;
    }
  }

  __syncthreads();
  if (tid < 32)
    mlp_wmma(sV0, sV1, w1, b1, w2, b2, w3, b3, vout, lane, bBase, Btot);
}

__global__ void zero_policy_kernel(float* __restrict__ p, int n)
{
  for (int i = blockIdx.x * blockDim.x + threadIdx.x; i < n;
       i += gridDim.x * blockDim.x)
    p[i] = 0.f;
}

extern "C" void kernel_launch(void* const* d_in, const int* in_sizes, int n_in,
                              void* d_out, int out_size, void* d_ws, size_t ws_size,
                              hipStream_t stream)
{
  const int*   board = (const int*)  d_in[0];
  const float* emb   = (const float*)d_in[1];
  const float* w1    = (const float*)d_in[2];
  const float* b1    = (const float*)d_in[3];
  const float* w2    = (const float*)d_in[4];
  const float* b2    = (const float*)d_in[5];
  const float* w3    = (const float*)d_in[6];
  const float* b3    = (const float*)d_in[7];
  float* out = (float*)d_out;

  const int B = in_sizes[0] / 450;            // board: (B, 2, 15, 15) int32
  const int nBlocks = (B + 15) / 16;
  const size_t qbytes = (size_t)in_sizes[1];  // int8 table = emb elements * 1 B

  if (ws_size >= qbytes) {
    // pre-quantize the table (8 MB, L2-resident), then gather int8 rows
    const int n4 = in_sizes[1] / 4;
    int qb = (n4 + 255) / 256; if (qb > 4096) qb = 4096;
    quant_emb_kernel<<<qb, 256, 0, stream>>>(emb, (unsigned*)d_ws, n4);
    nnue_fwd_q8<<<nBlocks, 128, 0, stream>>>(board, (const unsigned*)d_ws,
                                             w1, b1, w2, b2, w3, b3, out, B);
  } else {
    nnue_fwd_f32<<<nBlocks, 128, 0, stream>>>(board, emb,
                                              w1, b1, w2, b2, w3, b3, out, B);
  }

  float* pol = out + (size_t)B * 3;
  const int npol = B * 225;
  int zb = (npol + 255) / 256; if (zb > 2048) zb = 2048;
  zero_policy_kernel<<<zb, 256, 0, stream>>>(pol, npol);
}